// ATOCActorNet_70918499991639
// MI455X (gfx1250) — compile-verified
//
#include <hip/hip_runtime.h>
#include <math.h>

typedef float v2f __attribute__((ext_vector_type(2)));
typedef float v8f __attribute__((ext_vector_type(8)));

#define B_    128
#define N_    128
#define OBS_  256
#define D_    256
#define ACT_  64
#define ADIM_ 64
#define MG_   16
#define H_    128
#define ROWS_ (B_ * N_)

__device__ __forceinline__ v8f wmma_f32(v2f a, v2f b, v8f c) {
  // V_WMMA_F32_16X16X4_F32: D = A(16x4 f32) * B(4x16 f32) + C(16x16 f32)
  return __builtin_amdgcn_wmma_f32_16x16x4_f32(false, a, false, b, (short)0, c,
                                               false, false);
}

__device__ __forceinline__ float sigm(float x) { return 1.0f / (1.0f + __expf(-x)); }

// Async DMA: 16 bytes global -> LDS, tracked by ASYNCcnt (CDNA5 path).
// LDS offset = low 32 bits of the generic shared-memory pointer.
__device__ __forceinline__ void async_ld_b128(void* lds_dst, const float* gsrc) {
  unsigned loff = (unsigned)(size_t)lds_dst;
  asm volatile("global_load_async_to_lds_b128 %0, %1, off"
               :: "v"(loff), "v"(gsrc) : "memory");
}
__device__ __forceinline__ void wait_async0() {
  asm volatile("s_wait_asynccnt 0x0" ::: "memory");
}

// ---------------------------------------------------------------------------
// Generic Y = X @ W^T (+bias), optional ReLU applied to X, optional batching.
// Block: 256 threads (8 waves). Block tile: 64(M) x 32(N). K chunk: 32.
// Double-buffered LDS staging via global_load_async_to_lds_b128; each wave
// computes a 16x16 tile with a K-loop of v_wmma_f32_16x16x4_f32.
// Requires M%64==0, N%32==0, K%32==0.
// ---------------------------------------------------------------------------
__global__ __launch_bounds__(256)
void wmma_gemm_kernel(const float* __restrict__ X, const float* __restrict__ W,
                      const float* __restrict__ bias, float* __restrict__ Y,
                      int M, int Nc, int K, int relu_in,
                      long long bsX, long long bsW, long long bsY) {
  __shared__ float As[2][64 * 32];
  __shared__ float Ws[2][32 * 32];
  const int tid  = threadIdx.x;
  const int wv   = tid >> 5;
  const int lane = tid & 31;
  const int hlf  = lane >> 4;   // 0: K=0..1, 1: K=2..3 within a wmma step
  const int r    = lane & 15;
  const int wm   = wv & 3;      // wave M tile (0..3)
  const int wn   = wv >> 2;     // wave N tile (0..1)
  const int tileM = blockIdx.x * 64;
  const int tileN = blockIdx.y * 32;
  X += (long long)blockIdx.z * bsX;
  W += (long long)blockIdx.z * bsW;
  Y += (long long)blockIdx.z * bsY;

  auto stage = [&](int buf, int k0) {
    // A tile 64x32: 512 float4 -> 2 async B128 per thread
#pragma unroll
    for (int j = 0; j < 2; ++j) {
      int idx = j * 256 + tid;
      int row = idx >> 3, kq = idx & 7;
      async_ld_b128(&As[buf][row * 32 + kq * 4],
                    X + (long long)(tileM + row) * K + k0 + kq * 4);
    }
    // W tile 32(N)x32(K): 256 float4 -> 1 async B128 per thread
    {
      int row = tid >> 3, kq = tid & 7;
      async_ld_b128(&Ws[buf][row * 32 + kq * 4],
                    W + (long long)(tileN + row) * K + k0 + kq * 4);
    }
  };

  v8f acc = {};
  stage(0, 0);
  for (int k0 = 0; k0 < K; k0 += 32) {
    const int buf = (k0 >> 5) & 1;
    wait_async0();       // this wave's staged chunk landed in LDS
    __syncthreads();     // everyone's chunk landed; prior-buf readers done
    if (k0 + 32 < K) stage(buf ^ 1, k0 + 32);  // overlap next chunk's DMA
    const float* ap = &As[buf][(wm * 16 + r) * 32 + hlf * 2];
    const float* bp = &Ws[buf][(wn * 16 + r) * 32 + hlf * 2];
#pragma unroll
    for (int ks = 0; ks < 32; ks += 4) {
      v2f a; a.x = ap[ks]; a.y = ap[ks + 1];
      if (relu_in) { a.x = fmaxf(a.x, 0.f); a.y = fmaxf(a.y, 0.f); }
      v2f b; b.x = bp[ks]; b.y = bp[ks + 1];
      acc = wmma_f32(a, b, acc);
    }
  }
  // Epilogue: C/D layout -> m = vgpr + 8*(lane>=16), n = lane&15
  const int col  = tileN + wn * 16 + r;
  const float bb = bias ? bias[col] : 0.0f;
  const int mrow = tileM + wm * 16 + hlf * 8;
#pragma unroll
  for (int v = 0; v < 8; ++v) {
    Y[(long long)(mrow + v) * Nc + col] = acc[v] + bb;
  }
}

// ---------------------------------------------------------------------------
// LayerNorm (one wave32 per row), optional tanh epilogue. rows%8==0, cols%32==0.
// ---------------------------------------------------------------------------
__global__ __launch_bounds__(256)
void ln_kernel(const float* __restrict__ X, const float* __restrict__ g,
               const float* __restrict__ b, float* __restrict__ Y,
               int cols, int do_tanh) {
  const int row  = blockIdx.x * 8 + (threadIdx.x >> 5);
  const int lane = threadIdx.x & 31;
  const int npl  = cols >> 5;  // elements per lane (<= 8)
  const float* xr = X + (long long)row * cols;
  float vals[8];
  float s = 0.f, s2 = 0.f;
  for (int j = 0; j < npl; ++j) {
    float v = xr[j * 32 + lane];
    vals[j] = v; s += v; s2 += v * v;
  }
  for (int off = 16; off; off >>= 1) {
    s  += __shfl_xor(s, off, 32);
    s2 += __shfl_xor(s2, off, 32);
  }
  const float inv  = 1.0f / (float)cols;
  const float mu   = s * inv;
  const float var  = s2 * inv - mu * mu;
  const float rstd = rsqrtf(var + 1e-5f);
  float* yr = Y + (long long)row * cols;
  for (int j = 0; j < npl; ++j) {
    int c = j * 32 + lane;
    float o = (vals[j] - mu) * rstd * g[c] + b[c];
    if (do_tanh) o = tanhf(o);
    yr[c] = o;
  }
}

// ---------------------------------------------------------------------------
// is_init = sigmoid(relu(att2) @ w3 + b3) > 0.5  <=>  logit > 0
// ---------------------------------------------------------------------------
__global__ __launch_bounds__(256)
void init_mask_kernel(const float* __restrict__ att2, const float* __restrict__ w3,
                      const float* __restrict__ b3, int* __restrict__ is_init) {
  const int row  = blockIdx.x * 8 + (threadIdx.x >> 5);
  const int lane = threadIdx.x & 31;
  const float* xr = att2 + (long long)row * ADIM_;
  float v = fmaxf(xr[lane], 0.f) * w3[lane] +
            fmaxf(xr[32 + lane], 0.f) * w3[32 + lane];
  for (int off = 16; off; off >>= 1) v += __shfl_xor(v, off, 32);
  if (lane == 0) is_init[row] = (v + b3[0]) > 0.0f ? 1 : 0;
}

__global__ __launch_bounds__(256)
void sqnorm_kernel(const float* __restrict__ obs, float* __restrict__ sq) {
  const int row  = blockIdx.x * 8 + (threadIdx.x >> 5);
  const int lane = threadIdx.x & 31;
  const float* xr = obs + (long long)row * OBS_;
  float s = 0.f;
  for (int j = 0; j < 8; ++j) { float v = xr[j * 32 + lane]; s += v * v; }
  for (int off = 16; off; off >>= 1) s += __shfl_xor(s, off, 32);
  if (lane == 0) sq[row] = s;
}

// ---------------------------------------------------------------------------
// Top-16 nearest by d2 = |xi|^2+|xj|^2-2*G[i][j] (sqrt is monotone), indices
// returned sorted ascending (matching jnp.sort(argsort[:16])).
// One thread per (b,i).
// ---------------------------------------------------------------------------
__global__ __launch_bounds__(256)
void topk_kernel(const float* __restrict__ G, const float* __restrict__ sq,
                 int* __restrict__ gidx) {
  const int t = blockIdx.x * 256 + threadIdx.x;
  const int b = t >> 7, i = t & 127;
  const float* Gr = G + ((long long)b * N_ + i) * N_;
  const float* sb = sq + b * N_;
  const float sqi = sb[i];
  unsigned long long selLo = 0ull, selHi = 0ull;
  int chosen[16];
  for (int p = 0; p < 16; ++p) {
    float best = 3.4e38f; int bj = 0;
    for (int j = 0; j < 128; ++j) {
      bool used = (j < 64) ? ((selLo >> j) & 1ull) : ((selHi >> (j - 64)) & 1ull);
      if (!used) {
        float d = sqi + sb[j] - 2.0f * Gr[j];
        if (d < best) { best = d; bj = j; }  // strict < keeps lowest index on ties
      }
    }
    if (bj < 64) selLo |= 1ull << bj; else selHi |= 1ull << (bj - 64);
    chosen[p] = bj;
  }
  for (int p = 1; p < 16; ++p) {  // insertion sort ascending
    int key = chosen[p], q = p - 1;
    while (q >= 0 && chosen[q] > key) { chosen[q + 1] = chosen[q]; --q; }
    chosen[q + 1] = key;
  }
  for (int p = 0; p < 16; ++p) gidx[(long long)t * MG_ + p] = chosen[p];
}

// C[b,i,:] = one_hot scatter of group indices, masked by is_init.
__global__ void cmask_kernel(float* __restrict__ C, const int* __restrict__ gidx,
                             const int* __restrict__ is_init) {
  const long long bi = blockIdx.x;
  float* Cr = C + bi * N_;
  Cr[threadIdx.x] = 0.0f;
  __syncthreads();
  if (threadIdx.x < MG_ && is_init[bi]) Cr[gidx[bi * MG_ + threadIdx.x]] = 1.0f;
}

__global__ void gather_kernel(const float* __restrict__ thoughts,
                              const int* __restrict__ gidx,
                              float* __restrict__ xg, int agent) {
  const int b = blockIdx.x, tid = threadIdx.x;
  const int* gi = gidx + ((long long)b * N_ + agent) * MG_;
  for (int k = 0; k < MG_; ++k) {
    int idx = gi[k];
    xg[((long long)b * MG_ + k) * D_ + tid] =
        thoughts[((long long)b * N_ + idx) * D_ + tid];
  }
}

__global__ void scatter_kernel(float* __restrict__ thoughts,
                               const int* __restrict__ gidx,
                               const float* __restrict__ integ,
                               const int* __restrict__ is_init, int agent) {
  const int b = blockIdx.x, tid = threadIdx.x;
  if (!is_init[(long long)b * N_ + agent]) return;
  const int* gi = gidx + ((long long)b * N_ + agent) * MG_;
  for (int k = 0; k < MG_; ++k) {
    int idx = gi[k];
    thoughts[((long long)b * N_ + idx) * D_ + tid] =
        integ[((long long)b * MG_ + k) * D_ + tid];
  }
}

// ---------------------------------------------------------------------------
// Bi-LSTM over the 16-step group sequence. grid = (B/16, 2 directions).
// Block owns 16 batch rows; h/c kept in LDS across timesteps; 8 waves each
// compute four 16x16 gate tiles via WMMA per timestep (gates = x@wih^T+h@whh^T).
// Gate order i,f,g,o (PyTorch). Writes integ[b, k, dir*128 + h].
// ---------------------------------------------------------------------------
__global__ __launch_bounds__(256)
void lstm_kernel(const float* __restrict__ xg,
                 const float* __restrict__ wih_f, const float* __restrict__ whh_f,
                 const float* __restrict__ bih_f, const float* __restrict__ bhh_f,
                 const float* __restrict__ wih_r, const float* __restrict__ whh_r,
                 const float* __restrict__ bih_r, const float* __restrict__ bhh_r,
                 float* __restrict__ integ) {
  __shared__ float hbuf[16 * 128];
  __shared__ float cbuf[16 * 128];
  __shared__ float gbuf[16 * 512];
  const int dir = blockIdx.y;
  const float* wih = dir ? wih_r : wih_f;
  const float* whh = dir ? whh_r : whh_f;
  const float* bih = dir ? bih_r : bih_f;
  const float* bhh = dir ? bhh_r : bhh_f;
  const int row0 = blockIdx.x * 16;
  const int tid  = threadIdx.x;
  const int wv   = tid >> 5;
  const int lane = tid & 31;
  const int hlf  = lane >> 4;
  const int r    = lane & 15;
  const int nb   = (wv * 4) * 16 + r;  // base gate column for this wave's q=0 tile

  for (int idx = tid; idx < 16 * 128; idx += 256) { hbuf[idx] = 0.f; cbuf[idx] = 0.f; }
  __syncthreads();

  for (int t = 0; t < 16; ++t) {
    const int tt = dir ? (15 - t) : t;
    v8f acc0 = {}, acc1 = {}, acc2 = {}, acc3 = {};
    // x @ wih^T  (K = 256), A fragments straight from gathered global buffer
    const float* xrow = xg + ((long long)(row0 + r) * MG_ + tt) * D_;
    for (int k = 0; k < 256; k += 4) {
      v2f a; a.x = xrow[k + hlf * 2]; a.y = xrow[k + hlf * 2 + 1];
      const float* wp = wih + k + hlf * 2;
      v2f b0, b1, b2, b3;
      b0.x = wp[(nb +  0) * 256]; b0.y = wp[(nb +  0) * 256 + 1];
      b1.x = wp[(nb + 16) * 256]; b1.y = wp[(nb + 16) * 256 + 1];
      b2.x = wp[(nb + 32) * 256]; b2.y = wp[(nb + 32) * 256 + 1];
      b3.x = wp[(nb + 48) * 256]; b3.y = wp[(nb + 48) * 256 + 1];
      acc0 = wmma_f32(a, b0, acc0); acc1 = wmma_f32(a, b1, acc1);
      acc2 = wmma_f32(a, b2, acc2); acc3 = wmma_f32(a, b3, acc3);
    }
    // h @ whh^T  (K = 128), A fragments from LDS-resident hidden state
    const float* hrow = hbuf + r * 128;
    for (int k = 0; k < 128; k += 4) {
      v2f a; a.x = hrow[k + hlf * 2]; a.y = hrow[k + hlf * 2 + 1];
      const float* wp = whh + k + hlf * 2;
      v2f b0, b1, b2, b3;
      b0.x = wp[(nb +  0) * 128]; b0.y = wp[(nb +  0) * 128 + 1];
      b1.x = wp[(nb + 16) * 128]; b1.y = wp[(nb + 16) * 128 + 1];
      b2.x = wp[(nb + 32) * 128]; b2.y = wp[(nb + 32) * 128 + 1];
      b3.x = wp[(nb + 48) * 128]; b3.y = wp[(nb + 48) * 128 + 1];
      acc0 = wmma_f32(a, b0, acc0); acc1 = wmma_f32(a, b1, acc1);
      acc2 = wmma_f32(a, b2, acc2); acc3 = wmma_f32(a, b3, acc3);
    }
    // spill gates (+bias) to LDS in [m][512] layout
#pragma unroll
    for (int q = 0; q < 4; ++q) {
      const int ncol = nb + q * 16;
      v8f acc = (q == 0) ? acc0 : (q == 1) ? acc1 : (q == 2) ? acc2 : acc3;
      const float bs = bih[ncol] + bhh[ncol];
#pragma unroll
      for (int v = 0; v < 8; ++v) gbuf[(hlf * 8 + v) * 512 + ncol] = acc[v] + bs;
    }
    __syncthreads();
    // pointwise cell update
    for (int idx = tid; idx < 16 * 128; idx += 256) {
      const int m = idx >> 7, hh = idx & 127;
      const float* gr = gbuf + m * 512;
      const float ig = sigm(gr[hh]);
      const float fg = sigm(gr[128 + hh]);
      const float gg = tanhf(gr[256 + hh]);
      const float og = sigm(gr[384 + hh]);
      const float c  = fg * cbuf[idx] + ig * gg;
      const float hv = og * tanhf(c);
      cbuf[idx] = c; hbuf[idx] = hv;
      integ[((long long)(row0 + m) * MG_ + tt) * D_ + dir * H_ + hh] = hv;
    }
    __syncthreads();
  }
}

// ---------------------------------------------------------------------------
extern "C" void kernel_launch(void* const* d_in, const int* in_sizes, int n_in,
                              void* d_out, int out_size, void* d_ws, size_t ws_size,
                              hipStream_t stream) {
  const float* obs      = (const float*)d_in[0];
  const float* a1_w1    = (const float*)d_in[1];
  const float* a1_b1    = (const float*)d_in[2];
  const float* a1_ln1_g = (const float*)d_in[3];
  const float* a1_ln1_b = (const float*)d_in[4];
  const float* a1_w2    = (const float*)d_in[5];
  const float* a1_b2    = (const float*)d_in[6];
  const float* a1_ln2_g = (const float*)d_in[7];
  const float* a1_ln2_b = (const float*)d_in[8];
  const float* att_w1   = (const float*)d_in[9];
  const float* att_b1   = (const float*)d_in[10];
  const float* att_w2   = (const float*)d_in[11];
  const float* att_b2   = (const float*)d_in[12];
  const float* att_w3   = (const float*)d_in[13];
  const float* att_b3   = (const float*)d_in[14];
  const float* wih_f    = (const float*)d_in[15];
  const float* whh_f    = (const float*)d_in[16];
  const float* bih_f    = (const float*)d_in[17];
  const float* bhh_f    = (const float*)d_in[18];
  const float* wih_r    = (const float*)d_in[19];
  const float* whh_r    = (const float*)d_in[20];
  const float* bih_r    = (const float*)d_in[21];
  const float* bhh_r    = (const float*)d_in[22];
  const float* a2_w1    = (const float*)d_in[23];
  const float* a2_b1    = (const float*)d_in[24];
  const float* a2_ln1_g = (const float*)d_in[25];
  const float* a2_ln1_b = (const float*)d_in[26];
  const float* a2_w2    = (const float*)d_in[27];
  const float* a2_b2    = (const float*)d_in[28];
  const float* a2_ln2_g = (const float*)d_in[29];
  const float* a2_ln2_b = (const float*)d_in[30];
  (void)in_sizes; (void)n_in; (void)out_size; (void)ws_size;

  float* acts = (float*)d_out;                            // [B,N,ACT]
  float* Cmat = (float*)d_out + (size_t)ROWS_ * ACT_;     // [B,N,N]

  float* ws = (float*)d_ws;
  size_t off = 0;
  float* x1    = ws + off; off += (size_t)ROWS_ * D_;     // also reused as h2
  float* t0    = ws + off; off += (size_t)ROWS_ * D_;     // thoughts (updated)
  float* att1  = ws + off; off += (size_t)ROWS_ * ADIM_;  // reused as acts_raw
  float* att2  = ws + off; off += (size_t)ROWS_ * ADIM_;
  float* sq    = ws + off; off += (size_t)ROWS_;
  float* G     = ws + off; off += (size_t)B_ * N_ * N_;
  float* xg    = ws + off; off += (size_t)B_ * MG_ * D_;
  float* integ = ws + off; off += (size_t)B_ * MG_ * D_;
  int* is_init = (int*)(ws + off); off += (size_t)ROWS_;
  int* gidx    = (int*)(ws + off); off += (size_t)ROWS_ * MG_;

  const dim3 blk(256);

  // actor_1
  wmma_gemm_kernel<<<dim3(ROWS_ / 64, D_ / 32, 1), blk, 0, stream>>>(
      obs, a1_w1, a1_b1, x1, ROWS_, D_, OBS_, 0, 0, 0, 0);
  ln_kernel<<<ROWS_ / 8, blk, 0, stream>>>(x1, a1_ln1_g, a1_ln1_b, x1, D_, 0);
  wmma_gemm_kernel<<<dim3(ROWS_ / 64, D_ / 32, 1), blk, 0, stream>>>(
      x1, a1_w2, a1_b2, t0, ROWS_, D_, D_, 1, 0, 0, 0);
  ln_kernel<<<ROWS_ / 8, blk, 0, stream>>>(t0, a1_ln2_g, a1_ln2_b, t0, D_, 0);

  // attention unit -> initiator mask
  wmma_gemm_kernel<<<dim3(ROWS_ / 64, ADIM_ / 32, 1), blk, 0, stream>>>(
      t0, att_w1, att_b1, att1, ROWS_, ADIM_, D_, 0, 0, 0, 0);
  wmma_gemm_kernel<<<dim3(ROWS_ / 64, ADIM_ / 32, 1), blk, 0, stream>>>(
      att1, att_w2, att_b2, att2, ROWS_, ADIM_, ADIM_, 1, 0, 0, 0);
  init_mask_kernel<<<ROWS_ / 8, blk, 0, stream>>>(att2, att_w3, att_b3, is_init);

  // pairwise distances: gram GEMM + sq norms, then top-16 selection + C mask
  sqnorm_kernel<<<ROWS_ / 8, blk, 0, stream>>>(obs, sq);
  wmma_gemm_kernel<<<dim3(N_ / 64, N_ / 32, B_), blk, 0, stream>>>(
      obs, obs, nullptr, G, N_, N_, OBS_, 0,
      (long long)N_ * OBS_, (long long)N_ * OBS_, (long long)N_ * N_);
  topk_kernel<<<ROWS_ / 256, blk, 0, stream>>>(G, sq, gidx);
  cmask_kernel<<<ROWS_, dim3(128), 0, stream>>>(Cmat, gidx, is_init);

  // sequential agent scan: gather -> bi-LSTM -> masked scatter
  for (int i = 0; i < N_; ++i) {
    gather_kernel<<<B_, blk, 0, stream>>>(t0, gidx, xg, i);
    lstm_kernel<<<dim3(B_ / 16, 2), blk, 0, stream>>>(
        xg, wih_f, whh_f, bih_f, bhh_f, wih_r, whh_r, bih_r, bhh_r, integ);
    scatter_kernel<<<B_, blk, 0, stream>>>(t0, gidx, integ, is_init, i);
  }

  // actor_2
  wmma_gemm_kernel<<<dim3(ROWS_ / 64, D_ / 32, 1), blk, 0, stream>>>(
      t0, a2_w1, a2_b1, x1, ROWS_, D_, D_, 1, 0, 0, 0);
  ln_kernel<<<ROWS_ / 8, blk, 0, stream>>>(x1, a2_ln1_g, a2_ln1_b, x1, D_, 0);
  wmma_gemm_kernel<<<dim3(ROWS_ / 64, ACT_ / 32, 1), blk, 0, stream>>>(
      x1, a2_w2, a2_b2, att1, ROWS_, ACT_, D_, 0, 0, 0, 0);
  ln_kernel<<<ROWS_ / 8, blk, 0, stream>>>(att1, a2_ln2_g, a2_ln2_b, acts, ACT_, 1);
}